// Yolov6Loss_66511863546259
// MI455X (gfx1250) — compile-verified
//
#include <hip/hip_runtime.h>

// ---------------------------------------------------------------------------
// Yolov6 assigner pre-pass for MI455X (gfx1250, wave32):
//   out[0 .. 17.2M)   distances  (2048 x 8400)  -- via V_WMMA_F32_16X16X4_F32
//   out[17.2M..34.4M) overlaps   (16 x 128 x 8400 == 2048 x 8400)
//   out[34.4M..51.6M) gious      (2048 x 8400)
// Store-bandwidth bound (~206 MB of write-once output) -> nontemporal stores.
// ---------------------------------------------------------------------------

typedef __attribute__((ext_vector_type(2))) float v2f;
typedef __attribute__((ext_vector_type(8))) float v8f;

#define NROW   2048
#define NANC   8400
#define COLT   525                    // 8400 / 16
#define ROWT   128                    // 2048 / 16
#define PLANE  17203200LL             // 2048 * 8400

// ---------------------------------------------------------------------------
// distances via rank-4 WMMA:  d^2 = |g|^2 + |a|^2 - 2 g.a
//   A row m  = [ gx, gy, |g|^2, 1 ]          (16x4, f32)
//   B col n  = [ -2ax, -2ay, 1, |a|^2 ]^T    (4x16, f32)
// One wave computes one 16x16 tile with a single v_wmma_f32_16x16x4_f32.
// ISA 7.12.2 f32 layouts:
//   A: lane L(0-15): V0=K0,V1=K1 for M=L;  lane L(16-31): V0=K2,V1=K3 for M=L-16
//   B: mirrored over N;  D: M = vgpr + 8*(lane>>4), N = lane&15
// ---------------------------------------------------------------------------
__global__ __launch_bounds__(256) void dist_wmma_kernel(
    const float4* __restrict__ gt,     // (2048,4)
    const float4* __restrict__ anc,    // (8400,4)
    float* __restrict__ out)           // (2048,8400)
{
  const int lane = threadIdx.x & 31;
  const int wave = threadIdx.x >> 5;
  const int tile = blockIdx.x * 8 + wave;        // 8400 blocks * 8 waves = 67200
  const int rowTile = tile / COLT;
  const int colTile = tile - rowTile * COLT;
  const int hi  = lane >> 4;                     // 0: K=0,1 half   1: K=2,3 half
  const int idx = lane & 15;

  float4 g = gt[rowTile * 16 + idx];
  float4 a = anc[colTile * 16 + idx];
  float gx = 0.5f * (g.x + g.z), gy = 0.5f * (g.y + g.w);
  float ax = 0.5f * (a.x + a.z), ay = 0.5f * (a.y + a.w);
  float g2 = gx * gx + gy * gy;
  float a2 = ax * ax + ay * ay;

  v2f Af, Bf;
  Af.x = hi ? g2   : gx;           // K=0 / K=2
  Af.y = hi ? 1.0f : gy;           // K=1 / K=3
  Bf.x = hi ? 1.0f : -2.0f * ax;
  Bf.y = hi ? a2   : -2.0f * ay;

  v8f c = {};
  // (neg_a, A, neg_b, B, c_mod, C, reuse_a, reuse_b)
  v8f d = __builtin_amdgcn_wmma_f32_16x16x4_f32(
      false, Af, false, Bf, (short)0, c, false, false);

  const int nCol  = colTile * 16 + idx;
  const int mBase = rowTile * 16 + hi * 8;
#pragma unroll
  for (int r = 0; r < 8; ++r) {
    // raw v_sqrt_f32 (<=1 ulp): d^2 in [0, ~8e5], no denorm/overflow range
    // issues, so skip the IEEE refinement sequence the libm sqrt expands to.
    float v = __builtin_amdgcn_sqrtf(fmaxf(d[r], 0.0f));
    __builtin_nontemporal_store(v, out + (long long)(mBase + r) * NANC + nCol);
  }
}

// ---------------------------------------------------------------------------
// Fused overlaps (IoU vs preds, eps added to union) + gious (GIoU vs anchors).
// Grid: y = row in [0,2048), x*256+tid = column in [0,8400). gt box is
// block-uniform; preds/anchors stay hot in L2 across the 2048 rows.
// ---------------------------------------------------------------------------
__global__ __launch_bounds__(256) void iou_giou_kernel(
    const float4* __restrict__ gt,     // (2048,4)  == (16,128,4)
    const float4* __restrict__ pred,   // (16,8400,4)
    const float4* __restrict__ anc,    // (8400,4)
    float* __restrict__ ov_out,        // (2048,8400)
    float* __restrict__ gi_out)        // (2048,8400)
{
  const int n = blockIdx.x * 256 + threadIdx.x;
  if (n >= NANC) return;
  const int row = blockIdx.y;
  const int b   = row >> 7;            // row / 128

  const float4 g  = gt[row];
  const float  ga = (g.z - g.x) * (g.w - g.y);
  const long long o = (long long)row * NANC + n;

  // ---- overlaps: IoU(gt[b,m], pred[b,n]),  union + 1e-9 ----
  {
    float4 p  = pred[b * NANC + n];
    float  pa = (p.z - p.x) * (p.w - p.y);
    float ltx = fmaxf(g.x, p.x), lty = fmaxf(g.y, p.y);
    float rbx = fminf(g.z, p.z), rby = fminf(g.w, p.w);
    float w = fmaxf(rbx - ltx, 0.0f), h = fmaxf(rby - lty, 0.0f);
    float ov = w * h;
    float un = ga + pa - ov + 1e-9f;
    __builtin_nontemporal_store(ov / un, ov_out + o);
  }

  // ---- gious: GIoU(gt_flat[row], anchor[n]) ----
  {
    float4 a  = anc[n];
    float  aa = (a.z - a.x) * (a.w - a.y);
    float ltx = fmaxf(g.x, a.x), lty = fmaxf(g.y, a.y);
    float rbx = fminf(g.z, a.z), rby = fminf(g.w, a.w);
    float w = fmaxf(rbx - ltx, 0.0f), h = fmaxf(rby - lty, 0.0f);
    float ov = w * h;
    float un = fmaxf(ga + aa - ov, 1e-6f);
    float iou = ov / un;
    float eltx = fminf(g.x, a.x), elty = fminf(g.y, a.y);
    float erbx = fmaxf(g.z, a.z), erby = fmaxf(g.w, a.w);
    float ew = fmaxf(erbx - eltx, 0.0f), eh = fmaxf(erby - elty, 0.0f);
    float enc = fmaxf(ew * eh, 1e-6f);
    __builtin_nontemporal_store(iou - (enc - un) / enc, gi_out + o);
  }
}

extern "C" void kernel_launch(void* const* d_in, const int* in_sizes, int n_in,
                              void* d_out, int out_size, void* d_ws, size_t ws_size,
                              hipStream_t stream) {
  const float4* gt   = (const float4*)d_in[0];   // 16*128*4  floats
  const float4* pred = (const float4*)d_in[1];   // 16*8400*4 floats
  const float4* anc  = (const float4*)d_in[2];   // 8400*4    floats

  float* dist = (float*)d_out;
  float* ovl  = dist + PLANE;
  float* gio  = dist + 2 * PLANE;

  // 67200 tiles of 16x16, 8 waves (tiles) per 256-thread block -> 8400 blocks
  dist_wmma_kernel<<<8400, 256, 0, stream>>>(gt, anc, dist);

  dim3 grid((NANC + 255) / 256, NROW);           // 33 x 2048
  iou_giou_kernel<<<grid, 256, 0, stream>>>(gt, pred, anc, ovl, gio);
}